// HandPoseLegitimizeLayer_26285199852062
// MI455X (gfx1250) — compile-verified
//
#include <hip/hip_runtime.h>
#include <math.h>

#define EPSF 1e-6f
#define HPB  256          // hands per block (8 wave32 waves)
#define NJ   21
#define FPH  63           // floats per hand

// ---------------- CDNA5 async global<->LDS availability ----------------
#if defined(__has_builtin)
#  if __has_builtin(__builtin_amdgcn_global_load_async_to_lds_b128) && \
      __has_builtin(__builtin_amdgcn_global_load_async_to_lds_b32)  && \
      __has_builtin(__builtin_amdgcn_global_store_async_from_lds_b128) && \
      __has_builtin(__builtin_amdgcn_global_store_async_from_lds_b32)  && \
      __has_builtin(__builtin_amdgcn_s_wait_asynccnt)
#    define HAVE_ASYNC_LDS 1
#  endif
#endif
#ifndef HAVE_ASYNC_LDS
#  define HAVE_ASYNC_LDS 0
#endif

#if HAVE_ASYNC_LDS
typedef int v4i __attribute__((vector_size(16)));
typedef __attribute__((address_space(1))) v4i* g_v4i_p;   // global int4*
typedef __attribute__((address_space(3))) v4i* l_v4i_p;   // LDS int4*
typedef __attribute__((address_space(1))) int* g_i32_p;   // global int*
typedef __attribute__((address_space(3))) int* l_i32_p;   // LDS int*

__device__ __forceinline__ g_v4i_p gq4(const void* p) {
    return (g_v4i_p)(unsigned long long)(size_t)p;        // inttoptr -> as(1)
}
__device__ __forceinline__ l_v4i_p lq4(const void* p) {
    return (l_v4i_p)(unsigned int)(size_t)p;              // low 32 bits = LDS offset
}
__device__ __forceinline__ g_i32_p gq1(const void* p) {
    return (g_i32_p)(unsigned long long)(size_t)p;
}
__device__ __forceinline__ l_i32_p lq1(const void* p) {
    return (l_i32_p)(unsigned int)(size_t)p;
}
#endif

// ---------------- small vector/matrix helpers ----------------
struct V3 { float x, y, z; };
struct M3 { float m00,m01,m02,m10,m11,m12,m20,m21,m22; };

__device__ __forceinline__ V3 v3(float x, float y, float z) { V3 r; r.x=x; r.y=y; r.z=z; return r; }
__device__ __forceinline__ V3 operator+(V3 a, V3 b) { return v3(a.x+b.x, a.y+b.y, a.z+b.z); }
__device__ __forceinline__ V3 operator-(V3 a, V3 b) { return v3(a.x-b.x, a.y-b.y, a.z-b.z); }
__device__ __forceinline__ V3 operator*(V3 a, float s) { return v3(a.x*s, a.y*s, a.z*s); }
__device__ __forceinline__ float dot3(V3 a, V3 b) { return a.x*b.x + a.y*b.y + a.z*b.z; }
__device__ __forceinline__ V3 cross3(V3 a, V3 b) {
    return v3(a.y*b.z - a.z*b.y, a.z*b.x - a.x*b.z, a.x*b.y - a.y*b.x);
}
__device__ __forceinline__ V3 unit3(V3 a) {
    float n = sqrtf(dot3(a, a));
    return a * (1.0f / (n + EPSF));
}
__device__ __forceinline__ float clip_pm1(float x) {
    return fminf(1.0f - EPSF, fmaxf(-1.0f + EPSF, x));
}
__device__ __forceinline__ V3 mv3(const M3& R, V3 v) {
    return v3(R.m00*v.x + R.m01*v.y + R.m02*v.z,
              R.m10*v.x + R.m11*v.y + R.m12*v.z,
              R.m20*v.x + R.m21*v.y + R.m22*v.z);
}

// Rodrigues: R = I + s*K + (1-c)*K^2, axis re-normalized like the reference
__device__ __forceinline__ M3 rotmat_cs(V3 axis, float c, float s) {
    V3 a = unit3(axis);
    float t = 1.0f - c;
    M3 R;
    R.m00 = 1.0f - t*(a.y*a.y + a.z*a.z);
    R.m01 = -s*a.z + t*a.x*a.y;
    R.m02 =  s*a.y + t*a.x*a.z;
    R.m10 =  s*a.z + t*a.x*a.y;
    R.m11 = 1.0f - t*(a.x*a.x + a.z*a.z);
    R.m12 = -s*a.x + t*a.y*a.z;
    R.m20 = -s*a.y + t*a.x*a.z;
    R.m21 =  s*a.x + t*a.y*a.z;
    R.m22 = 1.0f - t*(a.x*a.x + a.y*a.y);
    return R;
}

__device__ __forceinline__ M3 getrot(float angleP, float angle, float flexRatio,
                                     V3 axis, V3 startv, V3 endv) {
    const float PIF = 3.1415926f;
    if (angle > PIF / 180.0f * 120.0f) angle = PIF - angle;
    float dif = fmaxf(angle - angleP, 0.0f) * flexRatio;
    float c = cosf(dif), s = sinf(dif);
    M3 r0 = rotmat_cs(axis, c,  s);
    M3 r1 = rotmat_cs(axis, c, -s);
    V3 v0 = unit3(mv3(r0, startv));
    V3 v1 = unit3(mv3(r1, startv));
    return (fabsf(dot3(v0, endv)) > fabsf(dot3(v1, endv))) ? r0 : r1;
}

// Smallest eigenvector of symmetric PSD 3x3 (trigonometric eigenvalue +
// cross-product eigenvector). Sign-ambiguous, but projections are sign-invariant.
__device__ __forceinline__ V3 smallest_evec(float a00, float a01, float a02,
                                            float a11, float a12, float a22) {
    float p1 = a01*a01 + a02*a02 + a12*a12;
    float q  = (a00 + a11 + a22) * (1.0f/3.0f);
    float b00 = a00 - q, b11 = a11 - q, b22 = a22 - q;
    float p2 = b00*b00 + b11*b11 + b22*b22 + 2.0f*p1;
    float p  = sqrtf(p2 * (1.0f/6.0f)) + 1e-30f;
    float ip = 1.0f / p;
    float c00 = b00*ip, c11 = b11*ip, c22 = b22*ip;
    float c01 = a01*ip, c02 = a02*ip, c12 = a12*ip;
    float detB = c00*(c11*c22 - c12*c12)
               - c01*(c01*c22 - c12*c02)
               + c02*(c01*c12 - c11*c02);
    float r   = fminf(1.0f, fmaxf(-1.0f, 0.5f*detB));
    float phi = acosf(r) * (1.0f/3.0f);
    float lmin = q + 2.0f*p*cosf(phi + 2.0943951023931953f); // + 2*pi/3
    V3 r0 = v3(a00 - lmin, a01, a02);
    V3 r1 = v3(a01, a11 - lmin, a12);
    V3 r2 = v3(a02, a12, a22 - lmin);
    V3 c0 = cross3(r0, r1), c1 = cross3(r0, r2), c2 = cross3(r1, r2);
    float d0 = dot3(c0, c0), d1 = dot3(c1, c1), d2 = dot3(c2, c2);
    V3 best = c0; float db = d0;
    if (d1 > db) { best = c1; db = d1; }
    if (d2 > db) { best = c2; db = d2; }
    if (db < 1e-30f) best = v3(1.0f, 0.0f, 0.0f);
    return best;
}

__device__ __forceinline__ V3 palm_norm(const V3 j[NJ], int i) {
    const int PA[4] = {1, 4, 10, 10};
    const int PB[4] = {4, 10, 7, 7};
    return unit3(cross3(j[PA[i]] - j[0], j[PB[i]] - j[0]));
}

__device__ __forceinline__ void planarize(V3 j[NJ]) {
    const int JI[5][4] = {{1,2,3,17},{4,5,6,18},{10,11,12,19},{7,8,9,20},{13,14,15,16}};
    #pragma unroll
    for (int f = 0; f < 5; ++f) {
        V3 p0 = j[JI[f][0]], p1 = j[JI[f][1]], p2 = j[JI[f][2]], p3 = j[JI[f][3]];
        V3 m = (p0 + p1 + p2 + p3) * 0.25f;
        V3 c0 = p0 - m, c1 = p1 - m, c2 = p2 - m, c3 = p3 - m;
        float a00 = c0.x*c0.x + c1.x*c1.x + c2.x*c2.x + c3.x*c3.x;
        float a01 = c0.x*c0.y + c1.x*c1.y + c2.x*c2.y + c3.x*c3.y;
        float a02 = c0.x*c0.z + c1.x*c1.z + c2.x*c2.z + c3.x*c3.z;
        float a11 = c0.y*c0.y + c1.y*c1.y + c2.y*c2.y + c3.y*c3.y;
        float a12 = c0.y*c0.z + c1.y*c1.z + c2.y*c2.z + c3.y*c3.z;
        float a22 = c0.z*c0.z + c1.z*c1.z + c2.z*c2.z + c3.z*c3.z;
        V3 vh = unit3(smallest_evec(a00, a01, a02, a11, a12, a22));
        float vd = -dot3(m, vh);
        #pragma unroll
        for (int k = 0; k < 4; ++k) {
            V3 p = j[JI[f][k]];
            float dist = dot3(p, vh) + vd;
            j[JI[f][k]] = p - vh * dist;
        }
    }
}

__device__ __forceinline__ void abduction(V3 j[NJ]) {
    const float angleP = 3.14159265358979323846f / 9.0f;
    const float rectify[4] = {0.189f, 0.1331f, -0.1491f, 0.0347f};
    const int mcpidx[4] = {1, 4, 10, 7};
    const int pipidx[4] = {2, 5, 11, 8};
    const int children[4][3] = {{2,3,17},{5,6,18},{11,12,19},{8,9,20}};
    #pragma unroll
    for (int i = 0; i < 4; ++i) {
        V3 pn    = palm_norm(j, i);
        V3 mcp   = j[mcpidx[i]];
        V3 pip   = j[pipidx[i]];
        V3 wrist = j[0];
        float vd   = -dot3(mcp, pn);
        float dist = dot3(pip, pn) + vd;
        V3 projpip = pip - pn * dist;
        V3 dmp = mcp - pip;
        float dis = sqrtf(dot3(dmp, dmp));
        V3 dpm = projpip - mcp;
        float flexRatio = sqrtf(dot3(dpm, dpm)) / (dis + EPSF);
        if (flexRatio < 0.3f) flexRatio = 0.0f;
        V3 wristmcp   = unit3(mcp - wrist);
        V3 mcpprojpip = unit3(projpip - mcp);
        V3 mcppip     = unit3(pip - mcp);
        bool overflex = acosf(clip_pm1(dot3(wristmcp, mcppip))) > (3.14f * 0.5f);
        M3 Rr = rotmat_cs(pn, cosf(rectify[i]), sinf(rectify[i]));
        V3 rw = mv3(Rr, wristmcp);
        float angle = acosf(clip_pm1(dot3(rw, mcpprojpip)));
        if (overflex) angle = 0.0f;
        M3 rot = getrot(angleP, angle, flexRatio, pn, mcpprojpip, wristmcp);
        V3 pivot = mcp;
        #pragma unroll
        for (int k = 0; k < 3; ++k) {
            int c = children[i][k];
            j[c] = mv3(rot, j[c] - pivot) + pivot;
        }
    }
}

__device__ __forceinline__ void plane_rot(V3 j[NJ]) {
    const float angleN = 3.14159265358979323846f / 9.0f;
    const int FI[4][3]  = {{1,2,3},{4,5,6},{10,11,12},{7,8,9}};
    const int MCPS_[4]  = {1, 4, 10, 7};
    const int CH[4][3]  = {{2,3,17},{5,6,18},{11,12,19},{8,9,20}}; // pivot, c1, c2
    #pragma unroll
    for (int i = 0; i < 4; ++i) {
        V3 a = j[FI[i][0]], b = j[FI[i][1]], c = j[FI[i][2]];
        V3 mcppip = unit3(b - a);
        V3 pipdip = unit3(c - b);
        bool maskline = fabsf(dot3(mcppip, pipdip)) > 0.95f;
        V3 calc = unit3(cross3(mcppip, pipdip));
        V3 wm   = unit3(j[MCPS_[i]] - j[0]);
        V3 pn   = palm_norm(j, i);
        V3 stdd = unit3(cross3(wm, pn));
        float angle = acosf(clip_pm1(dot3(calc, stdd)));
        if (maskline) angle = 0.0f;
        M3 rot = getrot(angleN, angle, 1.0f, mcppip, calc, stdd);
        V3 pivot = j[CH[i][0]];
        #pragma unroll
        for (int k = 1; k < 3; ++k) {
            int cc = CH[i][k];
            j[cc] = mv3(rot, j[cc] - pivot) + pivot;
        }
    }
}

// ---------------- kernel ----------------
__global__ void __launch_bounds__(HPB)
hand_legitimize_kernel(const float* __restrict__ in, float* __restrict__ out, int N) {
    __shared__ __align__(16) float smem[HPB * FPH]; // 64512 B, 63-dword stride -> conflict-free
    const int tid    = threadIdx.x;
    const int block0 = blockIdx.x * HPB;
    int nvalid = N - block0;
    if (nvalid <= 0) return;
    if (nvalid > HPB) nvalid = HPB;
    const int ndw = nvalid * FPH;
    const float* gin  = in  + (size_t)block0 * FPH;
    float*       gout = out + (size_t)block0 * FPH;

    // ---- stage in: global -> LDS (async, coalesced b128) ----
#if HAVE_ASYNC_LDS
    {
        const int n128 = ndw >> 2;
        for (int c = tid; c < n128; c += HPB)
            __builtin_amdgcn_global_load_async_to_lds_b128(
                gq4(gin + 4*c), lq4(smem + 4*c), 0, 0);
        for (int c = (n128 << 2) + tid; c < ndw; c += HPB)
            __builtin_amdgcn_global_load_async_to_lds_b32(
                gq1(gin + c), lq1(smem + c), 0, 0);
        __builtin_amdgcn_s_wait_asynccnt(0);
    }
#else
    for (int c = tid; c < ndw; c += HPB) smem[c] = gin[c];
#endif
    __syncthreads();

    // ---- one thread per hand, all 21 joints in registers ----
    if (tid < nvalid) {
        V3 j[NJ];
        const float* s = smem + tid * FPH;
        #pragma unroll
        for (int k = 0; k < NJ; ++k) j[k] = v3(s[3*k], s[3*k+1], s[3*k+2]);

        planarize(j);
        abduction(j);
        plane_rot(j);
        planarize(j);

        float* d = smem + tid * FPH;
        #pragma unroll
        for (int k = 0; k < NJ; ++k) {
            d[3*k] = j[k].x; d[3*k+1] = j[k].y; d[3*k+2] = j[k].z;
        }
    }
    __syncthreads();

    // ---- stage out: LDS -> global (async, coalesced b128) ----
#if HAVE_ASYNC_LDS
    {
        const int n128 = ndw >> 2;
        for (int c = tid; c < n128; c += HPB)
            __builtin_amdgcn_global_store_async_from_lds_b128(
                gq4(gout + 4*c), lq4(smem + 4*c), 0, 0);
        for (int c = (n128 << 2) + tid; c < ndw; c += HPB)
            __builtin_amdgcn_global_store_async_from_lds_b32(
                gq1(gout + c), lq1(smem + c), 0, 0);
        __builtin_amdgcn_s_wait_asynccnt(0);
    }
#else
    for (int c = tid; c < ndw; c += HPB) gout[c] = smem[c];
#endif
}

extern "C" void kernel_launch(void* const* d_in, const int* in_sizes, int n_in,
                              void* d_out, int out_size, void* d_ws, size_t ws_size,
                              hipStream_t stream) {
    (void)n_in; (void)out_size; (void)d_ws; (void)ws_size;
    const float* in = (const float*)d_in[0];
    float* out = (float*)d_out;
    const int N = in_sizes[0] / (NJ * 3);       // 524288 hands
    const int blocks = (N + HPB - 1) / HPB;     // 2048 blocks of 256 (8 waves)
    hipLaunchKernelGGL(hand_legitimize_kernel, dim3(blocks), dim3(HPB), 0, stream,
                       in, out, N);
}